// ProcessRewardModel_11441792877185
// MI455X (gfx1250) — compile-verified
//
#include <hip/hip_runtime.h>

// Problem sizes (fixed by the reference).
#define BQ 8
#define SQ 4096
#define DQ 1024
#define ROWS (BQ * SQ)          // 32768 rows of h

typedef float v2f __attribute__((ext_vector_type(2)));
typedef float v4f __attribute__((ext_vector_type(4)));
typedef float v8f __attribute__((ext_vector_type(8)));

// ---------------------------------------------------------------------------
// Kernel 1: fused stream.  For each 16-row tile of h (16 x 1024 fp32):
//   - stream the tile with non-temporal global_load_b128 (2 rows per instr),
//   - forward it to the h-copy output with non-temporal global_store_b128,
//   - stage it in a padded, wave-private LDS tile,
//   - accumulate scores with V_WMMA_F32_16X16X4_F32 (B = w chunk replicated
//     across all 16 columns, so every output column is the row dot product).
// h is read exactly once; traffic = 128 MB read + 128 MB write (roofline
// ~11 us at 23.3 TB/s).  The LDS tile is wave-private, and LDS ops from a
// wave execute in order, so only a compiler scheduling fence
// (__builtin_amdgcn_wave_barrier) is needed between the stage and the WMMA
// reads -- no block barrier, no storecnt drain: streaming stores overlap
// with the matrix pipe.
// ---------------------------------------------------------------------------
__global__ __launch_bounds__(256) void prm_score_copy_kernel(
    const float* __restrict__ h,        // (B*S, D)
    const float* __restrict__ w,        // (D,)
    const float* __restrict__ bias,     // (1,)
    float* __restrict__ h_out,          // (B*S, D) copy
    float* __restrict__ scores_ws)      // (B*S,) raw scores (+bias)
{
    __shared__ float w_lds[DQ];            // 4 KB
    __shared__ float tile[8][16][68];      // 8 waves x 16 rows x (64+4 pad) ~ 34 KB

    const int t    = threadIdx.x;
    const int lane = t & 31;
    const int wave = t >> 5;
    const int hi   = lane >> 4;     // half-wave: 0 or 1
    const int lo   = lane & 15;

    // Stage w into LDS once per block (256 threads x 4 floats).
    *(v4f*)&w_lds[t * 4] = *(const v4f*)(w + t * 4);
    __syncthreads();   // real cross-wave dependency: keep the block barrier

    const int rowbase = blockIdx.x * 128 + wave * 16;   // 256 blocks * 128 rows
    const float* hrow = h     + (size_t)rowbase * DQ;
    float*       orow = h_out + (size_t)rowbase * DQ;

    v8f acc = {};   // 16x16 f32 accumulator (8 VGPRs)

    for (int kc = 0; kc < 16; ++kc) {       // K chunks of 64
        const int kbase = kc * 64;

        // Load 16 rows x 64 cols: one b128 per lane covers 2 rows.
        // Streamed once -> non-temporal on both the load and the copy store.
        #pragma unroll
        for (int i = 0; i < 8; ++i) {
            const int r = i * 2 + hi;
            const int c = lo * 4;
            const size_t off = (size_t)r * DQ + kbase + c;
            v4f v = __builtin_nontemporal_load((const v4f*)(hrow + off));
            __builtin_nontemporal_store(v, (v4f*)(orow + off));   // h passthrough
            *(v4f*)&tile[wave][r][c] = v;   // stage for WMMA A-fragments
        }
        // Wave-private tile: HW keeps per-wave LDS ops in order; just stop the
        // compiler from reordering DS stores vs. the cross-lane DS reads.
        __builtin_amdgcn_wave_barrier();

        // 16 WMMA steps consume K=64 of this chunk.
        #pragma unroll
        for (int k4 = 0; k4 < 16; ++k4) {
            const int koff = k4 * 4 + 2 * hi;
            // A 16x4 fragment: lane (lo) = row M, half-wave picks K pair.
            v2f a = *(const v2f*)&tile[wave][lo][koff];
            // B 4x16 fragment: w chunk broadcast to all 16 columns.
            v2f bm = *(const v2f*)&w_lds[kbase + koff];
            acc = __builtin_amdgcn_wmma_f32_16x16x4_f32(
                /*neg_a=*/false, a, /*neg_b=*/false, bm,
                /*c_mod=*/(short)0, acc, /*reuse_a=*/false, /*reuse_b=*/false);
        }
        __builtin_amdgcn_wave_barrier();   // WAR fence before next stage pass
    }

    // Column N=0 of D holds the row dots: lane 0 -> M=0..7, lane 16 -> M=8..15.
    const float bv = bias[0];
    if (lane == 0) {
        #pragma unroll
        for (int m = 0; m < 8; ++m) scores_ws[rowbase + m] = acc[m] + bv;
    } else if (lane == 16) {
        #pragma unroll
        for (int m = 0; m < 8; ++m) scores_ws[rowbase + 8 + m] = acc[m] + bv;
    }
}

// ---------------------------------------------------------------------------
// Kernel 2: per-batch stable compaction.  One block (1024 threads) per batch,
// 4 positions per thread.  Exclusive prefix sum of the mask gives each masked
// position its destination rank; unmasked slots become -inf; counts written.
// ---------------------------------------------------------------------------
__global__ __launch_bounds__(1024) void prm_compact_kernel(
    const int* __restrict__ mask,         // (B,S) 0/1
    const float* __restrict__ scores_ws,  // (B,S)
    float* __restrict__ out_scores,       // (B,S)
    float* __restrict__ out_counts)       // (B,)
{
    __shared__ int scan[1024];

    const int b    = blockIdx.x;
    const int t    = threadIdx.x;
    const int base = b * SQ;
    const int s0   = t * 4;

    const int m0 = mask[base + s0 + 0];
    const int m1 = mask[base + s0 + 1];
    const int m2 = mask[base + s0 + 2];
    const int m3 = mask[base + s0 + 3];

    scan[t] = m0 + m1 + m2 + m3;
    __syncthreads();

    // Hillis-Steele inclusive scan over per-thread sums.
    for (int off = 1; off < 1024; off <<= 1) {
        int v   = scan[t];
        int add = (t >= off) ? scan[t - off] : 0;
        __syncthreads();
        scan[t] = v + add;
        __syncthreads();
    }

    const int excl  = (t == 0) ? 0 : scan[t - 1];
    const int total = scan[1023];

    // Fill the whole row with -inf first.
    const float NEG = -__builtin_huge_valf();
    out_scores[base + s0 + 0] = NEG;
    out_scores[base + s0 + 1] = NEG;
    out_scores[base + s0 + 2] = NEG;
    out_scores[base + s0 + 3] = NEG;
    __syncthreads();   // make fills visible before scatter

    // Scatter masked scores to their stable rank (all ranks < total).
    int r = excl;
    if (m0) { out_scores[base + r] = scores_ws[base + s0 + 0]; ++r; }
    if (m1) { out_scores[base + r] = scores_ws[base + s0 + 1]; ++r; }
    if (m2) { out_scores[base + r] = scores_ws[base + s0 + 2]; ++r; }
    if (m3) { out_scores[base + r] = scores_ws[base + s0 + 3]; ++r; }

    if (t == 0) out_counts[b] = (float)total;
}

// ---------------------------------------------------------------------------
// Launch: inputs per setup_inputs() order:
//   d_in[0]=input_ids (unused), d_in[1]=step_mask, d_in[2]=h,
//   d_in[3]=w_scorer, d_in[4]=b_scorer.
// Output layout: [step_scores B*S | step_counts B | h B*S*D].
// ---------------------------------------------------------------------------
extern "C" void kernel_launch(void* const* d_in, const int* in_sizes, int n_in,
                              void* d_out, int out_size, void* d_ws, size_t ws_size,
                              hipStream_t stream) {
    (void)in_sizes; (void)n_in; (void)out_size; (void)ws_size;

    const int*   mask = (const int*)d_in[1];
    const float* h    = (const float*)d_in[2];
    const float* w    = (const float*)d_in[3];
    const float* bias = (const float*)d_in[4];

    float* out        = (float*)d_out;
    float* out_scores = out;                       // B*S
    float* out_counts = out + (size_t)BQ * SQ;     // B
    float* out_h      = out_counts + BQ;           // B*S*D

    float* scores_ws  = (float*)d_ws;              // B*S floats scratch

    // 256 blocks x 256 threads (8 waves); each wave owns a 16-row tile.
    prm_score_copy_kernel<<<ROWS / 128, 256, 0, stream>>>(h, w, bias, out_h, scores_ws);
    // One block per batch row for the compaction.
    prm_compact_kernel<<<BQ, 1024, 0, stream>>>(mask, scores_ws, out_scores, out_counts);
}